// IPIter_37804302139730
// MI455X (gfx1250) — compile-verified
//
#include <hip/hip_runtime.h>
#include <math.h>

typedef __attribute__((ext_vector_type(2))) float v2f;
typedef __attribute__((ext_vector_type(8))) float v8f;

constexpr int IMG_H = 512;
constexpr int IMG_W = 512;
constexpr int NBATCH = 32;
constexpr int HWPX = IMG_H * IMG_W;

constexpr int RED_BLOCKS = 64;     // partial-reduction blocks per image

constexpr int TILE_R = 32;         // workgroup output region rows
constexpr int TILE_C = 64;         // workgroup output region cols
constexpr int HALO = 4;            // 9x9 conv halo
constexpr int LROWS = TILE_R + 2 * HALO;   // 40
constexpr int LCOLS = TILE_C + 2 * HALO;   // 72
constexpr int LSTRIDE = 74;        // even (8B-aligned b64 loads), conflict-free bank map

// -------- workspace layout (floats) --------
// [0,            2048)  per-(batch,block) partial sums of Dx
// [2048,         4096)  per-(batch,block) partial sums of Dx^2
// [4096,         4128)  reg[b]

// ---------------------------------------------------------------------------
// Pass 1: deterministic partial reduction of sum / sumsq over Dx = (Dvx, Dhx)
// ---------------------------------------------------------------------------
__global__ __launch_bounds__(256) void reduce_dx_kernel(
    const float* __restrict__ x, float* __restrict__ ws) {
  __shared__ float ssum[8], ssum2[8];
  const int b = blockIdx.y;
  const float* xb = x + (size_t)b * HWPX;
  float s = 0.f, s2 = 0.f;
  for (int i = blockIdx.x * 256 + threadIdx.x; i < HWPX; i += RED_BLOCKS * 256) {
    const int h = i >> 9;
    const int w = i & (IMG_W - 1);
    const float xc = xb[i];
    const float dv = ((h < IMG_H - 1) ? xb[i + IMG_W] : xc) - xc;  // edge pad -> 0
    const float dh = ((w < IMG_W - 1) ? xb[i + 1] : xc) - xc;
    s += dv + dh;
    s2 += dv * dv + dh * dh;
  }
  for (int off = 16; off; off >>= 1) {
    s += __shfl_down(s, off);
    s2 += __shfl_down(s2, off);
  }
  const int wave = threadIdx.x >> 5, lane = threadIdx.x & 31;
  if (lane == 0) { ssum[wave] = s; ssum2[wave] = s2; }
  __syncthreads();
  if (threadIdx.x == 0) {
    float t = 0.f, t2 = 0.f;
    for (int i = 0; i < 8; ++i) { t += ssum[i]; t2 += ssum2[i]; }
    ws[b * RED_BLOCKS + blockIdx.x] = t;
    ws[2048 + b * RED_BLOCKS + blockIdx.x] = t2;
  }
}

// ---------------------------------------------------------------------------
// Pass 2: fold partials -> reg[b] = reg_mul * std_approx[b] / (std + reg_const)
// ---------------------------------------------------------------------------
__global__ void compute_reg_kernel(const float* __restrict__ reg_mul,
                                   const float* __restrict__ reg_constant,
                                   const float* __restrict__ std_approx,
                                   float* __restrict__ ws) {
  const int b = threadIdx.x;
  if (b >= NBATCH) return;
  float s = 0.f, s2 = 0.f;
  for (int i = 0; i < RED_BLOCKS; ++i) {
    s += ws[b * RED_BLOCKS + i];
    s2 += ws[2048 + b * RED_BLOCKS + i];
  }
  const float n = 2.f * (float)HWPX;
  const float avg = s / n;
  const float var = fmaxf(s2 / n - avg * avg, 0.f);
  ws[4096 + b] = reg_mul[0] * std_approx[b] / (sqrtf(var) + reg_constant[0]);
}

// ---------------------------------------------------------------------------
// Pass 3: fused 9x9 WMMA conv + TV term + gradient step + Cardano prox
// Each wave32 computes one 16x16 output tile with 54x V_WMMA_F32_16X16X4_F32.
// ---------------------------------------------------------------------------
__global__ __launch_bounds__(256) void ipiter_main_kernel(
    const float* __restrict__ x, const float* __restrict__ htb,
    const float* __restrict__ hth, const float* __restrict__ gamma_p,
    const float* __restrict__ mu_p, const float* __restrict__ delta_p,
    const float* __restrict__ ws, float* __restrict__ out) {
  __shared__ float xs[LROWS * LSTRIDE];       // x tile, zero halo outside image
  __shared__ float bsm[9 * 6 * 64];           // banded B: [kh][chunk][n][k]

  const int tid = threadIdx.x;
  const int b = blockIdx.z;
  const int WR0 = blockIdx.y * TILE_R;
  const int WC0 = blockIdx.x * TILE_C;
  const float* xb = x + (size_t)b * HWPX;

  // Precompute the 54 banded 4x16 B matrices (same for every tile):
  // B[k][n] = w[kh, chunk*4 + k - n] if in [0,9) else 0, stored as [n][k].
  for (int e = tid; e < 9 * 6 * 64; e += 256) {
    const int kh = e / (6 * 64);
    const int rem = e % (6 * 64);
    const int chunk = rem >> 6;
    const int idx = rem & 63;
    const int n = idx >> 2, k = idx & 3;
    const int t = chunk * 4 + k - n;
    bsm[e] = (t >= 0 && t < 9) ? hth[kh * 9 + t] : 0.f;
  }
  // Stage x tile with zero padding outside the image (matches 'constant' pad).
  for (int e = tid; e < LROWS * LCOLS; e += 256) {
    const int r = e / LCOLS, c = e % LCOLS;
    const int gr = WR0 - HALO + r, gc = WC0 - HALO + c;
    float v = 0.f;
    if (gr >= 0 && gr < IMG_H && gc >= 0 && gc < IMG_W) v = xb[gr * IMG_W + gc];
    xs[r * LSTRIDE + c] = v;
  }
  __syncthreads();

  const int wave = tid >> 5, lane = tid & 31;
  const int g = lane >> 4, ln = lane & 15;        // lane half / lane-in-half
  const int m_off = (wave >> 2) * 16;             // 2 tile rows of waves
  const int n_off = (wave & 3) * 16;              // 4 tile cols of waves

  // A (16x4 fp32): lane L holds M = L%16, K = 2*(L/16)+{0,1} (consecutive cols)
  // B (4x16 fp32): lane L holds N = L%16, K = 2*(L/16)+{0,1} -> [n][k] contiguous
  v8f acc = {};
  const int arow0 = m_off + ln;
  const int acol = n_off + (g << 1);
  for (int kh = 0; kh < 9; ++kh) {
    const float* arow = &xs[(arow0 + kh) * LSTRIDE + acol];
    const float* brow = &bsm[kh * 6 * 64 + (ln << 2) + (g << 1)];
#pragma unroll
    for (int ch = 0; ch < 6; ++ch) {
      const v2f a = *(const v2f*)(arow + ch * 4);
      const v2f bb = *(const v2f*)(brow + ch * 64);
      acc = __builtin_amdgcn_wmma_f32_16x16x4_f32(
          /*neg_a=*/false, a, /*neg_b=*/false, bb,
          /*c_mod=*/(short)0, acc, /*reuse_a=*/false, /*reuse_b=*/false);
    }
  }

  // Elementwise epilogue: accumulator layout -> pixel (m = r + 8*g, n = ln).
  const float gamma = gamma_p[0];
  const float d = delta_p[0];
  const float inv_d2 = 1.f / (d * d);
  const float reg = ws[4096 + b];
  const float tprox = gamma * mu_p[b];
  const float* htbb = htb + (size_t)b * HWPX;
  float* outb = out + (size_t)b * HWPX;
  const float THIRD = 1.f / 3.f;
  const float T23 = 2.0943951023931953f;  // 2*pi/3

#pragma unroll
  for (int r = 0; r < 8; ++r) {
    const int m = r + 8 * g, n = ln;
    const int lr = m_off + m + HALO, lc = n_off + n + HALO;
    const int gr = WR0 + m_off + m, gc = WC0 + n_off + n;
    const float xc = xs[lr * LSTRIDE + lc];
    // forward differences with edge clamp (zero at far edge)
    const float dv = ((gr < IMG_H - 1) ? xs[(lr + 1) * LSTRIDE + lc] : xc) - xc;
    const float dh = ((gc < IMG_W - 1) ? xs[lr * LSTRIDE + lc + 1] : xc) - xc;
    // D^T terms: Dvx[h-1] - Dvx[h] with edge clamp at h=0
    const float pv = (gr > 0) ? (xc - xs[(lr - 1) * LSTRIDE + lc]) : dv;
    const float ph = (gc > 0) ? (xc - xs[lr * LSTRIDE + lc - 1]) : dh;
    const float dtd = (pv - dv + ph - dh) * inv_d2 *
                      rsqrtf((dv * dv + dh * dh) * inv_d2 + 1.f);
    const float grad = acc[r] - htbb[gr * IMG_W + gc] + reg * dtd;
    const float xt = xc - gamma * grad;
    // Cardano prox on [0,1]
    const float Bc = -(1.f + xt);
    const float Cc = xt - 2.f * tprox;
    const float Dc = tprox;
    const float p = Cc - Bc * Bc * THIRD;
    const float q = Dc - Bc * Cc * THIRD + 2.f * Bc * Bc * Bc * (1.f / 27.f);
    const float disc = 0.25f * q * q + p * p * p * (1.f / 27.f);
    const float sq = sqrtf(fmaxf(disc, 0.f));
    const float u1 = cbrtf(-0.5f * q + sq) + cbrtf(-0.5f * q - sq) - Bc * THIRD;
    const float p_safe = fminf(p, -1e-12f);
    const float mm = 2.f * sqrtf(-p_safe * THIRD);
    const float arg = fminf(fmaxf(3.f * q / (p_safe * mm), -1.f), 1.f);
    const float theta = acosf(arg) * THIRD;
    const float r0 = mm * cosf(theta) - Bc * THIRD;
    const float r1 = mm * cosf(theta - T23) - Bc * THIRD;
    const float r2 = mm * cosf(theta - 2.f * T23) - Bc * THIRD;
    const bool in0 = (r0 > 0.f) && (r0 < 1.f);
    const bool in1 = (r1 > 0.f) && (r1 < 1.f);
    const float u3 = in0 ? r0 : (in1 ? r1 : r2);
    const float u = (disc > 0.f) ? u1 : u3;
    outb[gr * IMG_W + gc] = fminf(fmaxf(u, 0.f), 1.f);
  }
}

// ---------------------------------------------------------------------------
extern "C" void kernel_launch(void* const* d_in, const int* in_sizes, int n_in,
                              void* d_out, int out_size, void* d_ws, size_t ws_size,
                              hipStream_t stream) {
  const float* gamma        = (const float*)d_in[0];
  const float* mu           = (const float*)d_in[1];
  const float* reg_mul      = (const float*)d_in[2];
  const float* reg_constant = (const float*)d_in[3];
  const float* delta        = (const float*)d_in[4];
  const float* x            = (const float*)d_in[5];
  const float* htb          = (const float*)d_in[6];
  const float* std_approx   = (const float*)d_in[7];
  const float* hth          = (const float*)d_in[12];
  float* out = (float*)d_out;
  float* ws = (float*)d_ws;

  reduce_dx_kernel<<<dim3(RED_BLOCKS, NBATCH), 256, 0, stream>>>(x, ws);
  compute_reg_kernel<<<1, 32, 0, stream>>>(reg_mul, reg_constant, std_approx, ws);
  ipiter_main_kernel<<<dim3(IMG_W / TILE_C, IMG_H / TILE_R, NBATCH), 256, 0, stream>>>(
      x, htb, hth, gamma, mu, delta, ws, out);
}